// GraphMoESwitch_10101763080599
// MI455X (gfx1250) — compile-verified
//
#include <hip/hip_runtime.h>
#include <hip/hip_bf16.h>

// ---------------------------------------------------------------------------
// GraphMoESwitch on MI455X (gfx1250): fp32 WMMA (V_WMMA_F32_16X16X4_F32)
// GraphConv rewritten as  segsum(h[src])@Wn  (weight factored out of edges),
// making the problem node-level GEMMs + one edge scatter per aggregation.
// Weight tiles staged into LDS by the Tensor Data Mover (tensor_load_to_lds,
// TENSORcnt) with TDM padding (64 dwords + 8 pad) producing a 72-dword row
// stride so the two half-wave B-fragment reads hit disjoint bank ranges
// (2*72 mod 64 = 16). Wave tile 32(M) x 64(N): each B fragment feeds 2 WMMAs.
// ---------------------------------------------------------------------------

#define NNODES 50000
#define NEDGES 800000
#define INDIM  6
#define HID    256
#define NEXP   8
#define NGRAPH 64
#define LDSW   72           // padded LDS row stride (dwords) for weight tiles

typedef __attribute__((ext_vector_type(2))) float v2f;
typedef __attribute__((ext_vector_type(8))) float v8f;
typedef __attribute__((ext_vector_type(4))) unsigned int u32x4;
typedef __attribute__((ext_vector_type(8))) int i32x8;
typedef __attribute__((ext_vector_type(4))) int i32x4;

#if defined(__gfx1250__) && __has_builtin(__builtin_amdgcn_tensor_load_to_lds)
#define USE_TDM 1
#else
#define USE_TDM 0
#endif

// ---------------------------------------------------------------- utilities
__global__ void zero_kernel(float* __restrict__ p, long long n) {
    long long i = (long long)blockIdx.x * blockDim.x + threadIdx.x;
    if (i < n) p[i] = 0.0f;
}

// h = relu(x @ W_enc + b_enc)   x:[N,6] W:[6,256]
__global__ void encoder_kernel(const float* __restrict__ x,
                               const float* __restrict__ W,
                               const float* __restrict__ b,
                               float* __restrict__ h) {
    int gid = blockIdx.x * blockDim.x + threadIdx.x;
    if (gid >= NNODES * HID) return;
    int i = gid >> 8, f = gid & 255;
    float s = b[f];
#pragma unroll
    for (int k = 0; k < INDIM; ++k) s += x[i * INDIM + k] * W[k * HID + f];
    h[gid] = fmaxf(s, 0.0f);
}

__global__ void node_count_kernel(const int* __restrict__ batch, float* __restrict__ cnt) {
    int i = blockIdx.x * blockDim.x + threadIdx.x;
    if (i < NNODES) atomicAdd(&cnt[batch[i]], 1.0f);
}

__global__ void edge_count_kernel(const int* __restrict__ src,
                                  const int* __restrict__ batch, float* __restrict__ cnt) {
    int e = blockIdx.x * blockDim.x + threadIdx.x;
    if (e < NEDGES) atomicAdd(&cnt[batch[src[e]]], 1.0f);
}

__global__ void size_feat_kernel(const int* __restrict__ batch,
                                 const float* __restrict__ ncnt,
                                 const float* __restrict__ ecnt,
                                 float* __restrict__ sf) {
    int i = blockIdx.x * blockDim.x + threadIdx.x;
    if (i >= NNODES) return;
    int g = batch[i];
    sf[i * 2 + 0] = log1pf(ncnt[g]);
    sf[i * 2 + 1] = log1pf(ecnt[g]);
}

// agg[dst] += val[src] over edges; 4 floats per thread (float4 gather + f32 atomics)
__global__ void scatter_kernel(const float* __restrict__ val,
                               const int* __restrict__ src,
                               const int* __restrict__ dst,
                               float* __restrict__ agg) {
    long long t = (long long)blockIdx.x * blockDim.x + threadIdx.x;
    if (t >= (long long)NEDGES * 64) return;
    int e = (int)(t >> 6);
    int f = (int)(t & 63) << 2;
    int s = src[e], d = dst[e];
    const float4 v = *(const float4*)(val + (size_t)s * HID + f);
    float* ap = agg + (size_t)d * HID + f;
    atomicAdd(ap + 0, v.x);
    atomicAdd(ap + 1, v.y);
    atomicAdd(ap + 2, v.z);
    atomicAdd(ap + 3, v.w);
}

// logits = r @ Wr2 + br2 ; idx = argmax (first-max tiebreak like jnp.argmax)
__global__ void router2_kernel(const float* __restrict__ r,
                               const float* __restrict__ W2,
                               const float* __restrict__ b2,
                               int* __restrict__ idx) {
    int i = blockIdx.x * blockDim.x + threadIdx.x;
    if (i >= NNODES) return;
    float acc[NEXP];
#pragma unroll
    for (int e = 0; e < NEXP; ++e) acc[e] = b2[e];
    const float* rp = r + (size_t)i * HID;
    for (int k = 0; k < HID; ++k) {
        float rv = rp[k];
#pragma unroll
        for (int e = 0; e < NEXP; ++e) acc[e] += rv * W2[k * NEXP + e];
    }
    int best = 0;
    float bv = acc[0];
#pragma unroll
    for (int e = 1; e < NEXP; ++e)
        if (acc[e] > bv) { bv = acc[e]; best = e; }
    idx[i] = best;
}

#if USE_TDM
// ---------------------------------------------------------------------------
// TDM: DMA a 256(K) x 64(col) f32 weight tile (row stride 256 elems in HBM)
// into LDS with a pad of 8 dwords after every 64 dwords -> LDSW=72 stride.
// D# layout per CDNA5 ISA §8.3/§8.4: group0 = {flags, lds_addr, gaddr_lo,
// gaddr_hi|type=2}; group1 packs data_size/pad config, tensor/tile dims,
// dim0 stride. Groups 2/3 + trailing group zero (2-D tensor). This toolchain
// exposes the 6-arg builtin: (u32x4, i32x8, i32x4, i32x4, i32x8, i32 cpol).
// Tracked by TENSORcnt (s_wait_tensorcnt).
// ---------------------------------------------------------------------------
__device__ __forceinline__ void tdm_load_tile(const float* gsrc, unsigned ldsByte) {
    const unsigned long long ga = (unsigned long long)(size_t)gsrc;
    u32x4 g0;
    g0[0] = 1u;                                                // count=1, user mode
    g0[1] = ldsByte;                                           // lds_addr
    g0[2] = (unsigned)(ga & 0xFFFFFFFFu);                      // global_addr[31:0]
    g0[3] = (unsigned)((ga >> 32) & 0x01FFFFFFu) | (2u << 30); // [56:32] | type=2
    const unsigned tdim0 = 1u << 20, tdim1 = 1u << 20;         // huge: no OOB clip
    const unsigned tile0 = 64u, tile1 = 256u, tile2 = 0u;
    const unsigned long long strd0 = 256ull;                   // elements per row
    i32x8 g1;
    g1[0] = (int)((2u << 16)      // data_size = 4 bytes
                | (1u << 20)      // pad_enable
                | (5u << 22)      // pad_interval: 64 dwords
                | (7u << 25));    // pad_amount:   8 dwords
    g1[1] = (int)((tdim0 & 0xFFFFu) << 16);
    g1[2] = (int)(((tdim0 >> 16) & 0xFFFFu) | ((tdim1 & 0xFFFFu) << 16));
    g1[3] = (int)(((tdim1 >> 16) & 0xFFFFu) | (tile0 << 16));
    g1[4] = (int)(tile1 | (tile2 << 16));
    g1[5] = (int)(strd0 & 0xFFFFFFFFull);
    g1[6] = (int)((strd0 >> 32) & 0xFFFFull);
    g1[7] = 0;
    i32x4 gz4 = {0, 0, 0, 0};
    i32x8 gz8 = {0, 0, 0, 0, 0, 0, 0, 0};
    __builtin_amdgcn_tensor_load_to_lds(g0, g1, gz4, gz4, gz8, 0);
}
#endif

// ---------------------------------------------------------------------------
// Fused dual-GEMM, N=K=256 fixed:  C = f(A1@W1 [+ A2@W2] + bias)
// MODE 0: router layer1: + sf rank-2 update from Wr1 rows 256/257, ReLU
// MODE 1: expert layer0: ReLU
// MODE 2: expert layer1: store row only where idx[row]==expert
// Tile: 8 waves/block; block tile 256(M) x 64(N); wave tile 32 x 64
// (2 M-subtiles -> each LDS B fragment feeds 2 WMMAs).
// V_WMMA_F32_16X16X4_F32 fragment layouts per CDNA5 ISA 7.12.2:
//   A 16x4: lanes 0-15 -> K=0,1 ; lanes 16-31 -> K=2,3 (M = lane&15)
//   B 4x16: vgpr j holds row K = 2*(lane>>4)+j, col N = lane&15
//   C 16x16: vgpr j -> row = 8*(lane>>4)+j, col = lane&15
// ---------------------------------------------------------------------------
template <int MODE>
__launch_bounds__(256)
__global__ void gemm256_kernel(const float* __restrict__ A1, const float* __restrict__ W1,
                               const float* __restrict__ A2, const float* __restrict__ W2,
                               const float* __restrict__ bias,
                               const float* __restrict__ sf, const float* __restrict__ Wsf,
                               const int* __restrict__ idx, int expert,
                               float* __restrict__ C, int M) {
    extern __shared__ float lds[];          // [ (MODE?2:1) * LDSW*HID ] floats
    const int lane   = threadIdx.x & 31;
    const int wave   = threadIdx.x >> 5;
    const int lhalf  = lane >> 4;   // 0: K pair 0,1 / C rows 0-7 ; 1: K pair 2,3 / rows 8-15
    const int lm     = lane & 15;   // M (for A) or N (for B/C) within tile
    const int rowBase = blockIdx.x * 256 + wave * 32;
    const int colBase = blockIdx.y * 64;

#if USE_TDM
    // ---- async weight-tile staging via Tensor Data Mover (one wave issues)
    if (wave == 0) {
        tdm_load_tile(W1 + colBase, (unsigned)(size_t)lds);
        if constexpr (MODE != 0)
            tdm_load_tile(W2 + colBase, (unsigned)(size_t)(lds + LDSW * HID));
        __builtin_amdgcn_s_wait_tensorcnt(0);
    }
#else
    // ---- cooperative staging fallback: K=256 rows x 64 cols, f32
    {
        const int tid = threadIdx.x;
#pragma unroll
        for (int i = 0; i < 16; ++i) {
            const int e = tid + (i << 8);   // 0..4095
            const int k = e >> 4;           // K row 0..255
            const int c = (e & 15) << 2;    // col 0,4,...,60
            *(float4*)(lds + k * LDSW + c) =
                *(const float4*)(W1 + (size_t)k * HID + colBase + c);
            if constexpr (MODE != 0) {
                *(float4*)(lds + LDSW * HID + k * LDSW + c) =
                    *(const float4*)(W2 + (size_t)k * HID + colBase + c);
            }
        }
    }
#endif

    v8f acc[8] = {v8f{}, v8f{}, v8f{}, v8f{}, v8f{}, v8f{}, v8f{}, v8f{}};

    int arow0 = rowBase + lm;
    int arow1 = rowBase + 16 + lm;
    if (arow0 >= M) arow0 = M - 1;          // clamp loads; stores masked in epilogue
    if (arow1 >= M) arow1 = M - 1;
    const float* a1p0 = A1 + (size_t)arow0 * HID + lhalf * 2;
    const float* a1p1 = A1 + (size_t)arow1 * HID + lhalf * 2;
    if constexpr (MODE != 0) {
        // warm second activation stream (global_prefetch_b8) while GEMM1 runs
        __builtin_prefetch((const void*)(A2 + (size_t)arow0 * HID), 0, 0);
        __builtin_prefetch((const void*)(A2 + (size_t)arow1 * HID), 0, 0);
    }

    __syncthreads();

    for (int kc = 0; kc < HID; kc += 4) {
        v2f a0 = *(const v2f*)(a1p0 + kc);
        v2f a1 = *(const v2f*)(a1p1 + kc);
        const int kb = kc + lhalf * 2;
        const float* lb = lds + kb * LDSW + lm;
#pragma unroll
        for (int f = 0; f < 4; ++f) {
            v2f b;
            b.x = lb[f * 16];           // row kb
            b.y = lb[f * 16 + LDSW];    // row kb+1
            acc[f] = __builtin_amdgcn_wmma_f32_16x16x4_f32(
                false, a0, false, b, (short)0, acc[f], false, false);
            acc[4 + f] = __builtin_amdgcn_wmma_f32_16x16x4_f32(
                false, a1, false, b, (short)0, acc[4 + f], false, false);
        }
    }
    if constexpr (MODE != 0) {
        const float* a2p0 = A2 + (size_t)arow0 * HID + lhalf * 2;
        const float* a2p1 = A2 + (size_t)arow1 * HID + lhalf * 2;
        const float* lds2 = lds + LDSW * HID;
        for (int kc = 0; kc < HID; kc += 4) {
            v2f a0 = *(const v2f*)(a2p0 + kc);
            v2f a1 = *(const v2f*)(a2p1 + kc);
            const int kb = kc + lhalf * 2;
            const float* lb = lds2 + kb * LDSW + lm;
#pragma unroll
            for (int f = 0; f < 4; ++f) {
                v2f b;
                b.x = lb[f * 16];
                b.y = lb[f * 16 + LDSW];
                acc[f] = __builtin_amdgcn_wmma_f32_16x16x4_f32(
                    false, a0, false, b, (short)0, acc[f], false, false);
                acc[4 + f] = __builtin_amdgcn_wmma_f32_16x16x4_f32(
                    false, a1, false, b, (short)0, acc[4 + f], false, false);
            }
        }
    }

    // epilogue
#pragma unroll
    for (int f = 0; f < 4; ++f) {
        const int n = colBase + f * 16 + lm;
        const float bv = bias[n];
        float w0 = 0.f, w1 = 0.f;
        if constexpr (MODE == 0) { w0 = Wsf[n]; w1 = Wsf[HID + n]; }
#pragma unroll
        for (int mi = 0; mi < 2; ++mi) {
#pragma unroll
            for (int j = 0; j < 8; ++j) {
                const int row = rowBase + mi * 16 + lhalf * 8 + j;
                if (row < M) {
                    float v = acc[mi * 4 + f][j] + bv;
                    if constexpr (MODE == 0)
                        v += sf[row * 2 + 0] * w0 + sf[row * 2 + 1] * w1;
                    if constexpr (MODE <= 1) v = fmaxf(v, 0.0f);
                    if constexpr (MODE == 2) {
                        if (idx[row] == expert) C[(size_t)row * HID + n] = v;
                    } else {
                        C[(size_t)row * HID + n] = v;
                    }
                }
            }
        }
    }
}

// ---------------------------------------------------------------------------
extern "C" void kernel_launch(void* const* d_in, const int* in_sizes, int n_in,
                              void* d_out, int out_size, void* d_ws, size_t ws_size,
                              hipStream_t stream) {
    (void)in_sizes; (void)n_in; (void)out_size; (void)ws_size;

    const float* x     = (const float*)d_in[0];
    const int*   edge  = (const int*)d_in[1];
    const int*   batch = (const int*)d_in[2];
    const float* W_enc = (const float*)d_in[3];
    const float* b_enc = (const float*)d_in[4];
    const float* Wr1   = (const float*)d_in[5];   // [258,256]
    const float* br1   = (const float*)d_in[6];
    const float* Wr2   = (const float*)d_in[7];   // [256,8]
    const float* br2   = (const float*)d_in[8];
    const float* Ws0   = (const float*)d_in[9];   // [8,256,256]
    const float* Wn0   = (const float*)d_in[10];
    const float* b0    = (const float*)d_in[11];  // [8,256]
    const float* Ws1   = (const float*)d_in[12];
    const float* Wn1   = (const float*)d_in[13];
    const float* b1    = (const float*)d_in[14];

    const int* src = edge;
    const int* dst = edge + NEDGES;

    // workspace layout (floats)
    float* ws    = (float*)d_ws;
    const size_t NH = (size_t)NNODES * HID;
    float* h     = ws;            // [N,256]
    float* agg   = h + NH;        // [N,256]  segsum(h[src])
    float* z     = agg + NH;      // [N,256]  router hidden, then expert hidden
    float* agg2  = z + NH;        // [N,256]  segsum(z[src]) per expert
    float* sf    = agg2 + NH;     // [N,2]
    float* cnts  = sf + (size_t)NNODES * 2;  // 64 node-cnt + 64 edge-cnt
    int*   idx   = (int*)(cnts + 2 * NGRAPH);
    float* out   = (float*)d_out;

    const dim3 blk(256);
    const dim3 ggrid((NNODES + 255) / 256, HID / 64);  // 196 x 4
    const int ldsOne = LDSW * HID * (int)sizeof(float);      //  73,728 B
    const int ldsTwo = 2 * LDSW * HID * (int)sizeof(float);  // 147,456 B

    // allow >64KB dynamic LDS (non-stream API; safe under graph capture)
    (void)hipFuncSetAttribute((const void*)gemm256_kernel<0>,
                              hipFuncAttributeMaxDynamicSharedMemorySize, ldsOne);
    (void)hipFuncSetAttribute((const void*)gemm256_kernel<1>,
                              hipFuncAttributeMaxDynamicSharedMemorySize, ldsTwo);
    (void)hipFuncSetAttribute((const void*)gemm256_kernel<2>,
                              hipFuncAttributeMaxDynamicSharedMemorySize, ldsTwo);

    // counters + layer-0 aggregation buffer to zero
    zero_kernel<<<1, 2 * NGRAPH, 0, stream>>>(cnts, 2 * NGRAPH);
    zero_kernel<<<(int)((NH + 255) / 256), blk, 0, stream>>>(agg, (long long)NH);

    // shared encoder
    encoder_kernel<<<(NNODES * HID + 255) / 256, blk, 0, stream>>>(x, W_enc, b_enc, h);

    // graph size features
    node_count_kernel<<<(NNODES + 255) / 256, blk, 0, stream>>>(batch, cnts);
    edge_count_kernel<<<(NEDGES + 255) / 256, blk, 0, stream>>>(src, batch, cnts + NGRAPH);
    size_feat_kernel<<<(NNODES + 255) / 256, blk, 0, stream>>>(batch, cnts, cnts + NGRAPH, sf);

    // router: z <- relu([h,sf] @ Wr1 + br1) ; idx <- argmax(z @ Wr2 + br2)
    gemm256_kernel<0><<<ggrid, blk, ldsOne, stream>>>(h, Wr1, nullptr, nullptr, br1,
                                                      sf, Wr1 + (size_t)HID * HID,
                                                      nullptr, 0, z, NNODES);
    router2_kernel<<<(NNODES + 255) / 256, blk, 0, stream>>>(z, Wr2, br2, idx);

    // expert-invariant neighbor aggregation of h
    const long long scatWork = (long long)NEDGES * 64;
    const int scatBlocks = (int)((scatWork + 255) / 256);
    scatter_kernel<<<scatBlocks, blk, 0, stream>>>(h, src, dst, agg);

    // dense compute-all-experts; output row stored by the argmax-selected expert
    for (int e = 0; e < NEXP; ++e) {
        const size_t wo = (size_t)e * HID * HID;
        const size_t bo = (size_t)e * HID;
        // layer 0: z = relu(h@Ws0[e] + agg@Wn0[e] + b0[e])
        gemm256_kernel<1><<<ggrid, blk, ldsTwo, stream>>>(h, Ws0 + wo, agg, Wn0 + wo, b0 + bo,
                                                          nullptr, nullptr, nullptr, 0, z, NNODES);
        // per-expert aggregation of z
        zero_kernel<<<(int)((NH + 255) / 256), blk, 0, stream>>>(agg2, (long long)NH);
        scatter_kernel<<<scatBlocks, blk, 0, stream>>>(z, src, dst, agg2);
        // layer 1: out[idx==e] = z@Ws1[e] + agg2@Wn1[e] + b1[e]
        gemm256_kernel<2><<<ggrid, blk, ldsTwo, stream>>>(z, Ws1 + wo, agg2, Wn1 + wo, b1 + bo,
                                                          nullptr, nullptr, idx, e, out, NNODES);
    }
}